// Attn_2130303779132
// MI455X (gfx1250) — compile-verified
//
#include <hip/hip_runtime.h>
#include <math.h>

#define S_LEN 4096
#define B_SZ  16
#define H_DIM 1024

typedef __attribute__((ext_vector_type(2))) float v2f;
typedef __attribute__((ext_vector_type(4))) float v4f;
typedef __attribute__((ext_vector_type(8))) float v8f;

// ---------------------------------------------------------------------------
// Kernel 1: v[b,n] = sum_k hidden[b,k] * W[k,n]   (v = hidden @ W)
// f32 WMMA 16x16x4. One wave per 16-wide N tile; grid = H/16 = 64 waves.
// A frag (16x4 f32, 2 VGPR): lanes 0-15 -> M=lane, K={k,k+1}; lanes 16-31 ->
// M=lane-16, K={k+2,k+3}. B frag mirrors with N=lane.
// ---------------------------------------------------------------------------
__global__ __launch_bounds__(32) void proj_hidden_wmma(
    const float* __restrict__ hidden,   // (B=16, H) row-major
    const float* __restrict__ W,        // (H, H) row-major: W[k*H + n]
    float* __restrict__ v_out) {        // (B, H)
  const int lane = threadIdx.x;
  const int mn   = lane & 15;           // M for A, N for B
  const int kh   = (lane >> 4) << 1;    // 0 or 2
  const int n0   = blockIdx.x * 16;

  v8f acc = {};
  for (int k = 0; k < H_DIM; k += 4) {
    v2f a, bm;
    // A: contiguous pair -> single b64 load
    const v2f* ap = (const v2f*)(hidden + mn * H_DIM + k + kh);
    a = *ap;
    // B: two rows of W, stride H
    bm.x = W[(size_t)(k + kh)     * H_DIM + n0 + mn];
    bm.y = W[(size_t)(k + kh + 1) * H_DIM + n0 + mn];
    acc = __builtin_amdgcn_wmma_f32_16x16x4_f32(
        /*neg_a=*/false, a, /*neg_b=*/false, bm,
        /*c_mod=*/(short)0, acc, /*reuse_a=*/false, /*reuse_b=*/false);
  }

  // D layout: VGPR r -> M = r (lanes 0-15) / r+8 (lanes 16-31); N = lane&15
  const int mbase = (lane >> 4) * 8;
#pragma unroll
  for (int r = 0; r < 8; ++r) {
    v_out[(mbase + r) * H_DIM + n0 + mn] = acc[r];
  }
}

// ---------------------------------------------------------------------------
// Kernel 2: E[b,s] = sum_h enc[s,b,h] * v[b,h]
// enc viewed as (S*B, H): wave i handles row i (b = i%16, s = i/16).
// Each lane streams 8 float4 (nontemporal: enc is touched exactly once),
// then a 5-step wave32 shuffle reduction. Fully coalesced 512B/instr.
// ---------------------------------------------------------------------------
__global__ __launch_bounds__(256) void energies_dot(
    const float* __restrict__ enc,      // (S*B, H)
    const float* __restrict__ v,        // (B, H) — hot in L2
    float* __restrict__ energ) {        // (B, S)
  const int wave = (int)((blockIdx.x * blockDim.x + threadIdx.x) >> 5);
  const int lane = threadIdx.x & 31;
  const int b = wave & (B_SZ - 1);
  const int s = wave >> 4;              // / B_SZ

  const v4f* erow = (const v4f*)(enc + (size_t)wave * H_DIM);
  const v4f* vrow = (const v4f*)(v + (size_t)b * H_DIM);

  float sum = 0.f;
#pragma unroll
  for (int c = 0; c < H_DIM / (4 * 32); ++c) {  // 8 chunks
    const int idx = c * 32 + lane;
    v4f e  = __builtin_nontemporal_load(erow + idx);
    v4f vv = vrow[idx];
    sum = fmaf(e.x, vv.x, sum);
    sum = fmaf(e.y, vv.y, sum);
    sum = fmaf(e.z, vv.z, sum);
    sum = fmaf(e.w, vv.w, sum);
  }
#pragma unroll
  for (int off = 16; off > 0; off >>= 1)
    sum += __shfl_down(sum, off, 32);
  if (lane == 0) energ[(size_t)b * S_LEN + s] = sum;
}

// ---------------------------------------------------------------------------
// Kernel 3: out[b,0,s] = softmax_s(E[b,s]). One 256-thread block per b;
// 16 elements/thread held in registers, two LDS tree reductions.
// ---------------------------------------------------------------------------
__global__ __launch_bounds__(256) void softmax_rows(
    const float* __restrict__ energ,    // (B, S)
    float* __restrict__ out) {          // (B, 1, S)
  const int b   = blockIdx.x;
  const int tid = threadIdx.x;
  __shared__ float red[256];

  float e[S_LEN / 256];
  float m = -INFINITY;
#pragma unroll
  for (int i = 0; i < S_LEN / 256; ++i) {
    e[i] = energ[(size_t)b * S_LEN + i * 256 + tid];
    m = fmaxf(m, e[i]);
  }
  red[tid] = m;
  __syncthreads();
  for (int off = 128; off > 0; off >>= 1) {
    if (tid < off) red[tid] = fmaxf(red[tid], red[tid + off]);
    __syncthreads();
  }
  m = red[0];
  __syncthreads();

  float ssum = 0.f;
#pragma unroll
  for (int i = 0; i < S_LEN / 256; ++i) {
    e[i] = expf(e[i] - m);
    ssum += e[i];
  }
  red[tid] = ssum;
  __syncthreads();
  for (int off = 128; off > 0; off >>= 1) {
    if (tid < off) red[tid] += red[tid + off];
    __syncthreads();
  }
  const float inv = 1.f / red[0];
#pragma unroll
  for (int i = 0; i < S_LEN / 256; ++i)
    out[(size_t)b * S_LEN + i * 256 + tid] = e[i] * inv;
}

// ---------------------------------------------------------------------------
extern "C" void kernel_launch(void* const* d_in, const int* in_sizes, int n_in,
                              void* d_out, int out_size, void* d_ws, size_t ws_size,
                              hipStream_t stream) {
  const float* hidden = (const float*)d_in[0];  // (1,B,H)
  const float* enc    = (const float*)d_in[1];  // (S,B,H)
  const float* W      = (const float*)d_in[2];  // (H,H)
  // d_in[3] = bias: constant offset per b under the softmax -> cancels; unused.
  (void)in_sizes; (void)n_in; (void)out_size; (void)ws_size;

  float* v_ws   = (float*)d_ws;                 // B*H floats   (64 KB)
  float* energ  = v_ws + B_SZ * H_DIM;          // B*S floats   (256 KB)
  float* out    = (float*)d_out;                // (B,1,S)

  proj_hidden_wmma<<<H_DIM / 16, 32, 0, stream>>>(hidden, W, v_ws);
  energies_dot<<<(S_LEN * B_SZ) / 8, 256, 0, stream>>>(enc, v_ws, energ);
  softmax_rows<<<B_SZ, 256, 0, stream>>>(energ, out);
}